// SparseLaplacianBuilder_66795331387714
// MI455X (gfx1250) — compile-verified
//
#include <hip/hip_runtime.h>
#include <hip/hip_bf16.h>
#include <stdint.h>

typedef __attribute__((ext_vector_type(2))) float v2f;
typedef __attribute__((ext_vector_type(8))) float v8f;

#define WAVE 32

// ---------------------------------------------------------------------------
// K0: init workspace (maps_diag = 0, block count = 1 (diag block), fill = 0)
// ---------------------------------------------------------------------------
__global__ void init_kernel(float* __restrict__ maps_diag, int* __restrict__ cnt,
                            int* __restrict__ fill, int n) {
  int i = blockIdx.x * blockDim.x + threadIdx.x;
  if (i < n * 16) maps_diag[i] = 0.0f;
  if (i < n) { cnt[i] = 1; fill[i] = 0; }
}

// ---------------------------------------------------------------------------
// K1: WMMA batched 4x4 products. One wave handles 4 edge-pairs per call.
//   A = [M0^T;M1^T;M2^T;M3^T] (16x4), B = [M0|M1|M2|M3] (4x16)
//   D = A*B holds Mi^T*Mj in block (i,j); we extract the i==j diagonal blocks.
// Per ISA A-layout (16x4 f32): lane (16h+L): {A[L][2h], A[L][2h+1]}
//   => lane (16h + 4g+p) holds {Mg[2h][p], Mg[2h+1][p]}.
// Assumed B-layout (4x16 f32): v0: lanes0-15=K0 row, lanes16-31=K2 row;
//   v1: K1/K3 => lane (16h + 4g+q) holds {Mg[2h][q], Mg[2h+1][q]}.
// Same per-lane elements for both operands of the same matrix => a == b.
// C/D layout: VGPR i, lanes0-15 -> (M=i, N=lane); lanes16-31 -> (M=8+i).
// ---------------------------------------------------------------------------
__global__ void wmma_pairs_kernel(const float* __restrict__ maps,
                                  const int* __restrict__ ei,   // (2, 2E) row-major
                                  float* __restrict__ maps_diag,
                                  float* __restrict__ triu,
                                  int E) {
  int wavesPerBlock = blockDim.x / WAVE;
  int wave = blockIdx.x * wavesPerBlock + (threadIdx.x / WAVE);
  int lane = threadIdx.x & (WAVE - 1);
  int quads = (E + 3) >> 2;
  if (wave >= quads) return;            // wave-uniform: EXEC stays all-ones

  int h   = lane >> 4;                  // lane half (row pair selector)
  int sub = lane & 15;
  int g   = sub >> 2;                   // which of the 4 matrices
  int q   = sub & 3;                    // column within the 4x4
  int e   = wave * 4 + g;               // this lane's pair-edge index
  bool ev = (e < E);

  v2f vL = {0.f, 0.f}, vR = {0.f, 0.f};
  if (ev) {
    const float* L = maps + (size_t)e * 16;         // left_maps[e]
    const float* R = maps + (size_t)(E + e) * 16;   // right_maps[e]
    int i0 = 8 * h + q;
    vL.x = L[i0]; vL.y = L[i0 + 4];
    vR.x = R[i0]; vR.y = R[i0 + 4];
  }
  v8f zero = {0.f, 0.f, 0.f, 0.f, 0.f, 0.f, 0.f, 0.f};
  v8f dLL = __builtin_amdgcn_wmma_f32_16x16x4_f32(false, vL, false, vL, (short)0, zero, false, false);
  v8f dRR = __builtin_amdgcn_wmma_f32_16x16x4_f32(false, vR, false, vR, (short)0, zero, false, false);
  v8f dLR = __builtin_amdgcn_wmma_f32_16x16x4_f32(false, vL, false, vR, (short)0, zero, false, false);

  // Diagonal block g lives in rows 4g..4g+3: lane half h must match (g>=2).
  bool active = ev && (h == (g >> 1));
  if (active) {
    float ll[4], rr[4], lr[4];
    if (g & 1) {
#pragma unroll
      for (int p = 0; p < 4; ++p) { ll[p] = dLL[p + 4]; rr[p] = dRR[p + 4]; lr[p] = dLR[p + 4]; }
    } else {
#pragma unroll
      for (int p = 0; p < 4; ++p) { ll[p] = dLL[p]; rr[p] = dRR[p]; lr[p] = dLR[p]; }
    }
    int nodeL = ei[e];        // edge_index[0][e]
    int nodeR = ei[E + e];    // edge_index[0][E+e]
#pragma unroll
    for (int p = 0; p < 4; ++p) {
      atomicAdd(&maps_diag[(size_t)nodeL * 16 + p * 4 + q], ll[p]);
      atomicAdd(&maps_diag[(size_t)nodeR * 16 + p * 4 + q], rr[p]);
      triu[(size_t)e * 16 + p * 4 + q] = lr[p];
    }
  }
}

// ---------------------------------------------------------------------------
// K2: per-node off-diagonal block counts (row e -> +1, col e -> +1)
// ---------------------------------------------------------------------------
__global__ void count_kernel(const int* __restrict__ ei, int* __restrict__ cnt,
                             int E, int E2) {
  int e = blockIdx.x * blockDim.x + threadIdx.x;
  if (e >= E) return;
  atomicAdd(&cnt[ei[e]], 1);        // row = edge_index[0][e]
  atomicAdd(&cnt[ei[E2 + e]], 1);   // col = edge_index[1][e]
}

// ---------------------------------------------------------------------------
// K3: single-workgroup exclusive scan (n = 50k -> 49 chunks of 1024)
// ---------------------------------------------------------------------------
__global__ void scan_kernel(const int* __restrict__ cnt, int* __restrict__ off, int n) {
  __shared__ int sdata[1024];
  __shared__ int carry;
  int tid = threadIdx.x;
  if (tid == 0) carry = 0;
  __syncthreads();
  for (int base = 0; base < n; base += 1024) {
    int i = base + tid;
    int v = (i < n) ? cnt[i] : 0;
    sdata[tid] = v;
    __syncthreads();
    for (int ofs = 1; ofs < 1024; ofs <<= 1) {
      int t = (tid >= ofs) ? sdata[tid - ofs] : 0;
      __syncthreads();
      sdata[tid] += t;
      __syncthreads();
    }
    int incl = sdata[tid];
    if (i < n) off[i] = carry + incl - v;   // exclusive
    __syncthreads();
    if (tid == 1023) carry += incl;         // chunk total
    __syncthreads();
  }
  if (tid == 0) off[n] = carry;
}

// ---------------------------------------------------------------------------
// K4: bucket 64-bit block keys per node: key = (col<<32) | stable_rank
//   rank: diag = 0, ij edge e -> e+1, ji edge e -> E+e+1  (matches JAX concat
//   order, so stable per-node sort reproduces the lexsort tie-breaking).
// ---------------------------------------------------------------------------
__global__ void scatter_diag_kernel(unsigned long long* __restrict__ keys,
                                    const int* __restrict__ off, int* __restrict__ fill, int n) {
  int i = blockIdx.x * blockDim.x + threadIdx.x;
  if (i >= n) return;
  int pos = off[i] + atomicAdd(&fill[i], 1);
  keys[pos] = ((unsigned long long)(unsigned)i << 32);   // col=i, rank=0
}

__global__ void scatter_edge_kernel(const int* __restrict__ ei,
                                    unsigned long long* __restrict__ keys,
                                    const int* __restrict__ off, int* __restrict__ fill,
                                    int E, int E2) {
  int e = blockIdx.x * blockDim.x + threadIdx.x;
  if (e >= E) return;
  int r = ei[e], c = ei[E2 + e];
  int p1 = off[r] + atomicAdd(&fill[r], 1);
  keys[p1] = ((unsigned long long)(unsigned)c << 32) | (unsigned)(e + 1);        // ij block
  int p2 = off[c] + atomicAdd(&fill[c], 1);
  keys[p2] = ((unsigned long long)(unsigned)r << 32) | (unsigned)(E + e + 1);    // ji block
}

// ---------------------------------------------------------------------------
// K5: per-node insertion sort of the tiny (avg 33) key list, in L2.
// ---------------------------------------------------------------------------
__global__ void sort_kernel(unsigned long long* __restrict__ keys,
                            const int* __restrict__ off, int n) {
  int i = blockIdx.x * blockDim.x + threadIdx.x;
  if (i >= n) return;
  int s = off[i], e = off[i + 1];
  for (int j = s + 1; j < e; ++j) {
    unsigned long long k = keys[j];
    int m = j - 1;
    while (m >= s && keys[m] > k) { keys[m + 1] = keys[m]; --m; }
    keys[m + 1] = k;
  }
}

// ---------------------------------------------------------------------------
// K6: emission. One wave per node. Entry positions are closed-form:
//   node region starts at 16*off[node]; sub-row a spans 4k entries; block t in
//   a run [rs,re) of equal cols emits at 4*rs + b*(re-rs) + (t-rs), which is
//   exactly the stable entry-level lexsort order (b-major within a dup run).
// ---------------------------------------------------------------------------
__global__ void emit_kernel(const unsigned long long* __restrict__ keys,
                            const int* __restrict__ off,
                            const float* __restrict__ maps_diag,
                            const float* __restrict__ triu,
                            float* __restrict__ out_i0,
                            float* __restrict__ out_i1,
                            float* __restrict__ out_val,
                            int n, int E) {
  int wavesPerBlock = blockDim.x / WAVE;
  int node = blockIdx.x * wavesPerBlock + (threadIdx.x / WAVE);
  int lane = threadIdx.x & (WAVE - 1);
  if (node >= n) return;
  int s = off[node], e = off[node + 1];
  int k = e - s;
  long long entryBase = 16LL * (long long)s;

  for (int t = lane; t < k; t += WAVE) {
    unsigned long long key = keys[s + t];
    int col = (int)(key >> 32);
    unsigned rank = (unsigned)key;
    // duplicate-(row,col) run bounds (runs are contiguous after sort)
    int rs = t;
    while (rs > 0 && (int)(keys[s + rs - 1] >> 32) == col) --rs;
    int re = t + 1;
    while (re < k && (int)(keys[s + re] >> 32) == col) ++re;
    int g = re - rs;

    float blk[16];
    if (rank == 0) {
#pragma unroll
      for (int j = 0; j < 16; ++j) blk[j] = maps_diag[(size_t)node * 16 + j];
    } else if (rank <= (unsigned)E) {
      size_t ed = (size_t)(rank - 1);
#pragma unroll
      for (int j = 0; j < 16; ++j) blk[j] = -triu[ed * 16 + j];
    } else {
      size_t ed = (size_t)(rank - 1u - (unsigned)E);
#pragma unroll
      for (int a = 0; a < 4; ++a)
#pragma unroll
        for (int b = 0; b < 4; ++b) blk[a * 4 + b] = -triu[ed * 16 + b * 4 + a];
    }
#pragma unroll
    for (int a = 0; a < 4; ++a) {
      long long rowBase = entryBase + (long long)a * 4 * k;
      float fi0 = (float)(node * 4 + a);
#pragma unroll
      for (int b = 0; b < 4; ++b) {
        long long p = rowBase + 4LL * rs + (long long)b * g + (t - rs);
        out_i0[p] = fi0;
        out_i1[p] = (float)(col * 4 + b);
        out_val[p] = blk[a * 4 + b];
      }
    }
  }
}

// ---------------------------------------------------------------------------
extern "C" void kernel_launch(void* const* d_in, const int* in_sizes, int n_in,
                              void* d_out, int out_size, void* d_ws, size_t ws_size,
                              hipStream_t stream) {
  const float* maps = (const float*)d_in[0];
  const int* ei = (const int*)d_in[1];
  int E2 = in_sizes[0] / 16;            // 2E = 1,600,000
  int E = E2 / 2;
  long long total = (long long)out_size / 3;      // n*16 + 2E*16 entries
  int n = (int)(total / 16) - E2;                 // 50,000

  // carve workspace (256B aligned)
  char* w = (char*)d_ws;
  auto carve = [&](size_t bytes) -> void* {
    void* p = (void*)w;
    w += (bytes + 255) & ~(size_t)255;
    return p;
  };
  float* maps_diag = (float*)carve((size_t)n * 16 * sizeof(float));
  float* triu      = (float*)carve((size_t)E * 16 * sizeof(float));
  int* cnt         = (int*)carve((size_t)n * sizeof(int));
  int* off         = (int*)carve(((size_t)n + 1) * sizeof(int));
  int* fill        = (int*)carve((size_t)n * sizeof(int));
  unsigned long long* keys =
      (unsigned long long*)carve(((size_t)n + 2 * (size_t)E) * sizeof(unsigned long long));
  (void)ws_size;

  const int T = 256;
  const int wpb = T / WAVE;

  init_kernel<<<(n * 16 + T - 1) / T, T, 0, stream>>>(maps_diag, cnt, fill, n);

  int quads = (E + 3) / 4;
  wmma_pairs_kernel<<<(quads + wpb - 1) / wpb, T, 0, stream>>>(maps, ei, maps_diag, triu, E);

  count_kernel<<<(E + T - 1) / T, T, 0, stream>>>(ei, cnt, E, E2);
  scan_kernel<<<1, 1024, 0, stream>>>(cnt, off, n);
  scatter_diag_kernel<<<(n + T - 1) / T, T, 0, stream>>>(keys, off, fill, n);
  scatter_edge_kernel<<<(E + T - 1) / T, T, 0, stream>>>(ei, keys, off, fill, E, E2);
  sort_kernel<<<(n + T - 1) / T, T, 0, stream>>>(keys, off, n);

  float* out = (float*)d_out;
  emit_kernel<<<(n + wpb - 1) / wpb, T, 0, stream>>>(keys, off, maps_diag, triu,
                                                     out, out + total, out + 2 * total,
                                                     n, E);
}